// Yolov1Loss_33071248179328
// MI455X (gfx1250) — compile-verified
//
#include <hip/hip_runtime.h>
#include <stdint.h>

#define THREADS 256
#define CELLS_PER_BLOCK 256
#define CH 30            // floats per cell
#define TILE_F (CELLS_PER_BLOCK * CH)       // 7680 floats per array
#define TILE_XF ((CELLS_PER_BLOCK * CH * 4) / 16)  // 1920 b128 transfers per array

// ---- CDNA5 async global -> LDS copy (ASYNCcnt tracked) ----
__device__ __forceinline__ void async_ld_b128(uint32_t lds_off, uint64_t gaddr) {
    asm volatile("global_load_async_to_lds_b128 %0, %1, off th:TH_LOAD_NT"
                 :: "v"(lds_off), "v"(gaddr) : "memory");
}
__device__ __forceinline__ void wait_async0() {
    asm volatile("s_wait_asynccnt 0" ::: "memory");
}
__device__ __forceinline__ uint32_t lds_off32(const void* p) {
    // low 32 bits of a generic LDS pointer == byte offset from workgroup LDS base
    return (uint32_t)(uintptr_t)p;
}

__global__ __launch_bounds__(THREADS)
void yolo_loss_partials(const float* __restrict__ inputs,
                        const float* __restrict__ targets,
                        float* __restrict__ partials,
                        int ncells) {
    __shared__ float lin[TILE_F];   // staged predictions
    __shared__ float ltg[TILE_F];   // staged targets
    __shared__ float red[THREADS];

    const int tid = threadIdx.x;
    const long long tile = (long long)blockIdx.x * CELLS_PER_BLOCK;
    int rem = ncells - (int)tile;
    if (rem > CELLS_PER_BLOCK) rem = CELLS_PER_BLOCK;
    const int nxf = (rem * (CH * 4) + 15) >> 4;   // b128 transfers per array

    const uint64_t gin = (uint64_t)(uintptr_t)(inputs  + tile * CH);
    const uint64_t gtg = (uint64_t)(uintptr_t)(targets + tile * CH);
    const uint32_t lin0 = lds_off32(lin);
    const uint32_t ltg0 = lds_off32(ltg);

    if (rem == CELLS_PER_BLOCK) {
        // full tile: 1920 transfers per array, straight-line issue
#pragma unroll
        for (int k = 0; k < 8; ++k) {
            int j = k * THREADS + tid;
            if (j < TILE_XF) async_ld_b128(lin0 + (uint32_t)j * 16u, gin + (uint64_t)j * 16u);
        }
#pragma unroll
        for (int k = 0; k < 8; ++k) {
            int j = k * THREADS + tid;
            if (j < TILE_XF) async_ld_b128(ltg0 + (uint32_t)j * 16u, gtg + (uint64_t)j * 16u);
        }
    } else {
        for (int j = tid; j < nxf; j += THREADS)
            async_ld_b128(lin0 + (uint32_t)j * 16u, gin + (uint64_t)j * 16u);
        for (int j = tid; j < nxf; j += THREADS)
            async_ld_b128(ltg0 + (uint32_t)j * 16u, gtg + (uint64_t)j * 16u);
    }
    wait_async0();      // this wave's async transfers done
    __syncthreads();    // all waves' transfers done -> LDS tile complete

    float cell_loss = 0.0f;
    if (tid < rem) {
        float pv[CH], tv[CH];
        const float2* lp = (const float2*)(lin + tid * CH);  // 120B stride, 8B aligned
        const float2* lt = (const float2*)(ltg + tid * CH);
#pragma unroll
        for (int i = 0; i < CH / 2; ++i) {
            float2 a = lp[i]; pv[2 * i] = a.x; pv[2 * i + 1] = a.y;
            float2 b = lt[i]; tv[2 * i] = b.x; tv[2 * i + 1] = b.y;
        }

        const float ig = 1.0f / 7.0f;
        const float cm = (tv[4] > 0.0f) ? 1.0f : 0.0f;
        const float nm = 1.0f - cm;

        // target box 0 corners
        const float tcx = tv[0] * ig, tcy = tv[1] * ig;
        const float thw = 0.5f * tv[2], thh = 0.5f * tv[3];
        const float tx0 = tcx - thw, ty0 = tcy - thh;
        const float tx1 = tcx + thw, ty1 = tcy + thh;
        const float area_t = (tx1 - tx0) * (ty1 - ty0);

        float iou[2];
#pragma unroll
        for (int b = 0; b < 2; ++b) {
            const float* pb = &pv[5 * b];
            float pcx = pb[0] * ig, pcy = pb[1] * ig;
            float phw = 0.5f * pb[2], phh = 0.5f * pb[3];
            float px0 = pcx - phw, py0 = pcy - phh;
            float px1 = pcx + phw, py1 = pcy + phh;
            float iw = fminf(px1, tx1) - fmaxf(px0, tx0);
            float ih = fminf(py1, ty1) - fmaxf(py0, ty0);
            iw = fmaxf(iw, 0.0f); ih = fmaxf(ih, 0.0f);
            float inter = iw * ih;
            float area_p = (px1 - px0) * (py1 - py0);
            float uni = fmaxf(area_p + area_t - inter, 1e-12f);
            iou[b] = inter / uni;
        }
        const bool r1 = iou[1] > iou[0];       // argmax, first wins on tie
        const float miou = fmaxf(iou[0], iou[1]);

        const float prx = r1 ? pv[5] : pv[0], pry = r1 ? pv[6] : pv[1];
        const float prw = r1 ? pv[7] : pv[2], prh = r1 ? pv[8] : pv[3];
        const float prc = r1 ? pv[9] : pv[4];
        const float trx = r1 ? tv[5] : tv[0], trY = r1 ? tv[6] : tv[1];
        const float trw = r1 ? tv[7] : tv[2], trh = r1 ? tv[8] : tv[3];

        const float dx = prx - trx, dy = pry - trY;
        const float lxy = dx * dx + dy * dy;

        const bool obj = cm > 0.0f;
        const float spw = obj ? prw : 1.0f, stw = obj ? trw : 1.0f;
        const float sph = obj ? prh : 1.0f, sth = obj ? trh : 1.0f;
        const float dw = sqrtf(spw) - sqrtf(stw);
        const float dh = sqrtf(sph) - sqrtf(sth);
        const float lwh = dw * dw + dh * dh;

        const float dob = prc - miou;
        const float lobj = dob * dob;

        const float d4 = pv[4] - tv[4], d9 = pv[9] - tv[9];
        const float lno = d4 * d4 + d9 * d9;

        float lcls = 0.0f;
#pragma unroll
        for (int i = 10; i < CH; ++i) {
            float d = pv[i] - tv[i];
            lcls += d * d;
        }

        cell_loss = cm * (0.5f * (lxy + lwh) + lobj + lcls) + 0.5f * nm * lno;
    }

    // deterministic block tree-reduction
    red[tid] = cell_loss;
    __syncthreads();
#pragma unroll
    for (int s = THREADS / 2; s > 0; s >>= 1) {
        if (tid < s) red[tid] += red[tid + s];
        __syncthreads();
    }
    if (tid == 0) partials[blockIdx.x] = red[0];
}

__global__ __launch_bounds__(256)
void yolo_reduce(const float* __restrict__ partials, int n,
                 float* __restrict__ out, float scale) {
    __shared__ float red[256];
    float s = 0.0f;
    for (int i = threadIdx.x; i < n; i += 256) s += partials[i];
    red[threadIdx.x] = s;
    __syncthreads();
#pragma unroll
    for (int k = 128; k > 0; k >>= 1) {
        if ((int)threadIdx.x < k) red[threadIdx.x] += red[threadIdx.x + k];
        __syncthreads();
    }
    if (threadIdx.x == 0) out[0] = red[0] * scale;
}

extern "C" void kernel_launch(void* const* d_in, const int* in_sizes, int n_in,
                              void* d_out, int out_size, void* d_ws, size_t ws_size,
                              hipStream_t stream) {
    (void)n_in; (void)out_size; (void)ws_size;
    const float* inputs  = (const float*)d_in[0];
    const float* targets = (const float*)d_in[1];
    float* out      = (float*)d_out;
    float* partials = (float*)d_ws;

    const int ncells  = in_sizes[0] / CH;          // 1,605,632
    const int nblocks = (ncells + CELLS_PER_BLOCK - 1) / CELLS_PER_BLOCK;  // 6272
    const int batch   = ncells / 49;               // 32768
    const float scale = 1.0f / (float)batch;

    yolo_loss_partials<<<nblocks, THREADS, 0, stream>>>(inputs, targets, partials, ncells);
    yolo_reduce<<<1, 256, 0, stream>>>(partials, nblocks, out, scale);
}